// Encoder_50379966382835
// MI455X (gfx1250) — compile-verified
//
#include <hip/hip_runtime.h>
#include <hip/hip_bf16.h>

// ---------------------------------------------------------------------------
// GCN encoder for MI455X (gfx1250, wave32).
// - GEMMs: V_WMMA_F32_16X16X4_F32, 32-row M-tile (N=100000 = 3125*32, so the
//   GEMM is fully branch-free; EXEC stays all-1s as WMMA requires).
// - X tile staged to LDS via the Tensor Data Mover (TENSOR_LOAD_TO_LDS with a
//   D# descriptor; LDS padding 128->130 DWORDs done by TDM pad fields),
//   falling back to a cooperative vector load if the builtin is absent.
// - Edge aggregation is the HBM roofline term (~5 GB traffic across 3 layers).
// ---------------------------------------------------------------------------

typedef float v2f  __attribute__((ext_vector_type(2)));
typedef float v8f  __attribute__((ext_vector_type(8)));
typedef unsigned int u32x4 __attribute__((ext_vector_type(4)));
typedef int   i32x4 __attribute__((ext_vector_type(4)));
typedef int   i32x8 __attribute__((ext_vector_type(8)));

#define HDIM 128
#define MT   32    // GEMM row tile (divides N exactly)
#define LSTR 130   // LDS row stride in floats (pad 128 -> 130: no bank conflicts)

#if __has_builtin(__builtin_amdgcn_tensor_load_to_lds) && \
    __has_builtin(__builtin_amdgcn_s_wait_tensorcnt)
#define USE_TDM 1
#else
#define USE_TDM 0
#endif

// --------------------------- utility kernels -------------------------------

__global__ void fill_kernel(float* __restrict__ p, float v, int n) {
    int i = blockIdx.x * blockDim.x + threadIdx.x;
    if (i < n) p[i] = v;
}

__global__ void deg_kernel(const int* __restrict__ dst, float* __restrict__ deg) {
    int i = blockIdx.x * blockDim.x + threadIdx.x;     // grid sized exactly to E
    atomicAdd(&deg[dst[i]], 1.0f);
}

__global__ void dinv_kernel(float* __restrict__ deg, int n) {
    int i = blockIdx.x * blockDim.x + threadIdx.x;
    if (i < n) deg[i] = rsqrtf(deg[i] + 1.0f);         // deg of A+I on dst
}

// --------------------------- WMMA GEMM -------------------------------------
// H = X @ W  (X:[n,128] fp32, W:[128,128] fp32, H:[n,128] fp32), n % 32 == 0.
// Block = 256 threads = 8 wave32s, owns a 32x128 X tile in LDS.
// Wave w computes 32x16 of output (cols [16w,16w+16)) with 2 accumulators,
// reusing each B fragment for 2 WMMAs. 64 WMMA per wave, zero branches.
__global__ void gemm_wmma_kernel(const float* __restrict__ X,
                                 const float* __restrict__ W,
                                 float* __restrict__ Hout) {
    __shared__ float Xs[MT * LSTR];

    const int mt  = blockIdx.x;
    const int tid = threadIdx.x;
    const float* __restrict__ Xbase = X + (size_t)mt * MT * HDIM;

#if USE_TDM
    // ---- Tensor Data Mover: DMA the 128x32 f32 tile into LDS with padding.
    // pad_interval=6 (pad after every 128 DWORDs = one row), pad_amount=1
    // (2 DWORDs) -> LDS row stride 130 DWORDs = LSTR.
    if (tid < 32) {
        unsigned long long gaddr = (unsigned long long)(uintptr_t)Xbase;
        unsigned int       laddr = (unsigned int)(uintptr_t)(void*)Xs; // LDS byte offset

        u32x4 g0;
        g0.x = 1u;                                   // count=1, user descriptor
        g0.y = laddr;                                // lds_addr [63:32]
        g0.z = (unsigned int)(gaddr & 0xffffffffu);  // global_addr [95:64]
        g0.w = (unsigned int)((gaddr >> 32) & 0x1ffffffu) | (2u << 30); // [120:96] | type=2

        i32x8 g1;
        g1[0] = (2 << 16)        // data_size = 4 bytes
              | (1 << 20)        // pad_enable
              | (6 << 22)        // pad_interval: 2^(6+1)=128 DWORDs
              | (1 << 25);       // pad_amount: 2 DWORDs
        g1[1] = (HDIM & 0xffff) << 16;   // tensor_dim0[15:0] = 128 (bits 63:48)
        g1[2] = (MT & 0xffff) << 16;     // tensor_dim0 hi=0; tensor_dim1[15:0]=32 (bits 95:80)
        g1[3] = (HDIM & 0xffff) << 16;   // tensor_dim1 hi=0; tile_dim0=128 (bits 127:112)
        g1[4] = (MT & 0xffff);           // tile_dim1 = 32; tile_dim2 = 0
        g1[5] = HDIM;                    // tensor_dim0_stride[31:0] = 128
        g1[6] = 0;                       // stride hi, tensor_dim1_stride lo
        g1[7] = 0;

        i32x4 gz4 = {0, 0, 0, 0};                    // groups 2/3: unused (2D tile)
        i32x8 gz8 = {0, 0, 0, 0, 0, 0, 0, 0};        // extra group (6-arg form)
        __builtin_amdgcn_tensor_load_to_lds(g0, g1, gz4, gz4, gz8, 0);
        __builtin_amdgcn_s_wait_tensorcnt(0);
    }
    __syncthreads();
#else
    // ---- Fallback: cooperative load, 4096 elements / 256 threads.
    #pragma unroll
    for (int i = 0; i < 16; ++i) {
        int e = i * 256 + tid;
        Xs[(e >> 7) * LSTR + (e & 127)] = Xbase[e];
    }
    __syncthreads();
#endif

    const int wave = tid >> 5;       // 0..7 -> column tile
    const int lane = tid & 31;
    const int half = lane >> 4;      // 0: K pair {0,1}; 1: K pair {2,3}
    const int m    = lane & 15;
    const int colB = wave * 16 + m;

    v8f acc0 = {};
    v8f acc1 = {};
    #pragma unroll 8
    for (int k0 = 0; k0 < HDIM; k0 += 4) {
        const int kb = k0 + half * 2;
        v2f b, a0, a1;
        // B 4x16 fragment: row K striped over lanes
        b.x = W[(size_t)kb * HDIM + colB];
        b.y = W[(size_t)(kb + 1) * HDIM + colB];
        // A 16x4 fragments (ISA 7.12.2): lanes 0-15 K=0,1; lanes 16-31 K=2,3
        a0.x = Xs[m * LSTR + kb];
        a0.y = Xs[m * LSTR + kb + 1];
        a1.x = Xs[(m + 16) * LSTR + kb];
        a1.y = Xs[(m + 16) * LSTR + kb + 1];
        acc0 = __builtin_amdgcn_wmma_f32_16x16x4_f32(false, a0, false, b,
                                                     (short)0, acc0, false, false);
        acc1 = __builtin_amdgcn_wmma_f32_16x16x4_f32(false, a1, false, b,
                                                     (short)0, acc1, false, false);
    }

    // D 16x16 fp32: VGPR r -> row r (lanes 0-15) / row r+8 (lanes 16-31)
    float* __restrict__ Hbase = Hout + (size_t)mt * MT * HDIM;
    #pragma unroll
    for (int r = 0; r < 8; ++r) {
        int row = r + half * 8;
        Hbase[(size_t)row * HDIM + colB]        = acc0[r];
        Hbase[(size_t)(row + 16) * HDIM + colB] = acc1[r];
    }
}

// --------------------------- edge aggregation ------------------------------
// One wave per edge: 32 lanes x float4 = 128 channels. Grid sized exactly.
// agg[dst] += h[src] * dinv[src]*dinv[dst]
__global__ void agg_kernel(const float* __restrict__ h,
                           const int* __restrict__ src,
                           const int* __restrict__ dst,
                           const float* __restrict__ dinv,
                           float* __restrict__ agg) {
    int gid  = blockIdx.x * blockDim.x + threadIdx.x;
    int e    = gid >> 5;
    int lane = gid & 31;
    int s = src[e];
    int d = dst[e];
    float coef = dinv[s] * dinv[d];
    const float4* hs = (const float4*)(h + (size_t)s * HDIM);
    float4 v = hs[lane];                       // global_load_b128
    float* ap = agg + (size_t)d * HDIM + lane * 4;
    atomicAdd(ap + 0, v.x * coef);
    atomicAdd(ap + 1, v.y * coef);
    atomicAdd(ap + 2, v.z * coef);
    atomicAdd(ap + 3, v.w * coef);
}

// ------------------- self-loop + bias + ReLU + BN stats --------------------
__global__ void relu_stats_kernel(float* __restrict__ z,          // agg, in-place
                                  const float* __restrict__ h,
                                  const float* __restrict__ dinv,
                                  const float* __restrict__ bias,
                                  float* __restrict__ stats,      // [2*HDIM]
                                  int n, int chunk) {
    const int c  = threadIdx.x;                 // 0..127
    const int n0 = blockIdx.x * chunk;
    const int n1 = min(n, n0 + chunk);
    const float bc = bias[c];
    float s = 0.0f, s2 = 0.0f;
    for (int nn = n0; nn < n1; ++nn) {
        float di = dinv[nn];
        size_t idx = (size_t)nn * HDIM + c;
        float v = z[idx] + h[idx] * di * di + bc;
        v = fmaxf(v, 0.0f);
        z[idx] = v;
        s  += v;
        s2 += v * v;
    }
    atomicAdd(&stats[c], s);
    atomicAdd(&stats[HDIM + c], s2);
}

__global__ void finalize_stats_kernel(float* __restrict__ stats, int n) {
    int c = threadIdx.x;                        // 128 threads, 1 block
    float inv_n = 1.0f / (float)n;
    float mu  = stats[c] * inv_n;
    float var = stats[HDIM + c] * inv_n - mu * mu;  // biased (training BN)
    stats[c]        = mu;
    stats[HDIM + c] = rsqrtf(var + 1e-5f);
}

// ---------------- BN apply + write h_cat + sum-pool g_cat ------------------
__global__ void bn_apply_kernel(const float* __restrict__ z,
                                const float* __restrict__ stats,
                                const float* __restrict__ g,
                                const float* __restrict__ beta,
                                const int* __restrict__ batch,
                                float* __restrict__ norm_out,   // next layer input
                                float* __restrict__ hcat,       // [n, 384]
                                float* __restrict__ gcat,       // [G, 384]
                                int layer, int n, int chunk) {
    const int c  = threadIdx.x;
    const float mu   = stats[c];
    const float rstd = stats[HDIM + c];
    const float gc   = g[c];
    const float bc   = beta[c];
    const int n0 = blockIdx.x * chunk;
    const int n1 = min(n, n0 + chunk);
    const int coff = layer * HDIM + c;
    for (int nn = n0; nn < n1; ++nn) {
        float y = (z[(size_t)nn * HDIM + c] - mu) * rstd * gc + bc;
        norm_out[(size_t)nn * HDIM + c] = y;
        hcat[(size_t)nn * 384 + coff]   = y;
        atomicAdd(&gcat[(size_t)batch[nn] * 384 + coff], y);
    }
}

// ---------------------------------------------------------------------------

extern "C" void kernel_launch(void* const* d_in, const int* in_sizes, int n_in,
                              void* d_out, int out_size, void* d_ws, size_t ws_size,
                              hipStream_t stream) {
    const int N_ = 100000;
    const int E_ = 1600000;
    const int G_ = 128;

    const float* x     = (const float*)d_in[0];
    const int*   ei    = (const int*)d_in[1];
    const int*   batch = (const int*)d_in[2];
    const float* Wl[3]    = {(const float*)d_in[3],  (const float*)d_in[7],  (const float*)d_in[11]};
    const float* bl[3]    = {(const float*)d_in[4],  (const float*)d_in[8],  (const float*)d_in[12]};
    const float* gl[3]    = {(const float*)d_in[5],  (const float*)d_in[9],  (const float*)d_in[13]};
    const float* betal[3] = {(const float*)d_in[6],  (const float*)d_in[10], (const float*)d_in[14]};

    const int* src = ei;            // edge_index[0,:]
    const int* dst = ei + E_;       // edge_index[1,:]

    // Workspace (floats): dinv[N] | P[N*128] | Q[N*128] | R[N*128] | stats[256]
    float* ws    = (float*)d_ws;
    float* dinv  = ws;
    float* P     = ws + N_;                         // normalized output / next input
    float* Q     = P + (size_t)N_ * HDIM;           // h = X @ W
    float* R     = Q + (size_t)N_ * HDIM;           // agg, then z (in-place)
    float* stats = R + (size_t)N_ * HDIM;           // [sum(128), sumsq(128)]

    float* out  = (float*)d_out;                    // h_cat [N,384]
    float* gcat = out + (size_t)N_ * 384;           // g_cat [G,384]

    // Zero accumulation targets (d_out/d_ws are poisoned, not re-zeroed).
    fill_kernel<<<(G_ * 384 + 255) / 256, 256, 0, stream>>>(gcat, 0.0f, G_ * 384);
    fill_kernel<<<(N_ + 255) / 256, 256, 0, stream>>>(dinv, 0.0f, N_);

    // Degree (A + I) on dst, then dinv = rsqrt(deg)
    deg_kernel<<<E_ / 256, 256, 0, stream>>>(dst, dinv);          // E % 256 == 0
    dinv_kernel<<<(N_ + 255) / 256, 256, 0, stream>>>(dinv, N_);

    const int mblocks = N_ / MT;                    // 3125, exact
    const int SBLOCKS = 512;
    const int chunk   = (N_ + SBLOCKS - 1) / SBLOCKS;

    const float* in = x;
    for (int l = 0; l < 3; ++l) {
        // h = in @ W   (WMMA fp32)
        gemm_wmma_kernel<<<mblocks, 256, 0, stream>>>(in, Wl[l], Q);
        // zero agg buffer + stats (contiguous)
        fill_kernel<<<((N_ * HDIM + 2 * HDIM) + 255) / 256, 256, 0, stream>>>(
            R, 0.0f, N_ * HDIM + 2 * HDIM);
        // agg[dst] += h[src] * dinv[src]*dinv[dst]   (one wave per edge)
        agg_kernel<<<(E_ * 32) / 256, 256, 0, stream>>>(Q, src, dst, dinv, R);
        // z = relu(agg + h*dinv^2 + b); per-channel stats
        relu_stats_kernel<<<SBLOCKS, 128, 0, stream>>>(R, Q, dinv, bl[l], stats, N_, chunk);
        finalize_stats_kernel<<<1, 128, 0, stream>>>(stats, N_);
        // y = (z - mu)*rstd*g + beta; write h_cat slice, pool into g_cat
        bn_apply_kernel<<<SBLOCKS, 128, 0, stream>>>(R, stats, gl[l], betal[l], batch,
                                                     P, out, gcat, l, N_, chunk);
        in = P;
    }
}